// CombineLoss_13477607375450
// MI455X (gfx1250) — compile-verified
//
#include <hip/hip_runtime.h>
#include <math.h>

#define B_SZ   512
#define HH     112
#define WW     112
#define HW     (HH*WW)        // 12544
#define NV4    (HW/4)         // 3136 float4 per plane
#define THREADS 256

typedef __attribute__((ext_vector_type(2))) float v2f;
typedef __attribute__((ext_vector_type(8))) float v8f;

__device__ __forceinline__ float lse2(float a, float b) {
    float m = fmaxf(a, b);
    return m + logf(expf(a - m) + expf(b - m));
}

// One block per sample b. Streams channel-1 CAM planes with float4 loads and
// accumulates the two squared-diff sums in V_WMMA_F32_16X16X4_F32 accumulators
// (A = packed squared diffs, B = ones => D accumulates row sums).
__global__ __launch_bounds__(THREADS)
void CombineLoss_cam_sums(const float* __restrict__ preds1,
                          const float* __restrict__ cams1,
                          const float* __restrict__ cams2,
                          const int*   __restrict__ y,
                          const int*   __restrict__ d_index,
                          float* __restrict__ S_er,
                          float* __restrict__ S_same)
{
    const int b   = blockIdx.x;
    const int tid = threadIdx.x;
    const int idx = (d_index[0] != 0) ? 1 : 0;
    const int yb  = y[b];

    if (yb == 0) {                       // er and same terms both masked out
        if (tid == 0) { S_er[b] = 0.f; S_same[b] = 0.f; }
        return;
    }

    // argmax flags (tie -> index 0, matching jnp.argmax)
    const float p10 = preds1[((size_t)idx      * B_SZ + b) * 2 + 0];
    const float p11 = preds1[((size_t)idx      * B_SZ + b) * 2 + 1];
    const float q0  = preds1[((size_t)(1 - idx)* B_SZ + b) * 2 + 0];
    const float q1  = preds1[((size_t)(1 - idx)* B_SZ + b) * 2 + 1];
    const int  cur       = (p11 > p10) ? 1 : 0;
    const int  flag      = (q1  > q0 ) ? 1 : 0;
    const bool need_same = (cur == flag);    // same_loss mask for this sample

    const float4* c1 = (const float4*)(cams1 + ((size_t)(idx      * B_SZ + b) * 2 + 1) * HW);
    const float4* c2 = (const float4*)(cams2 + ((size_t)(idx      * B_SZ + b) * 2 + 1) * HW);
    const float4* co = (const float4*)(cams1 + ((size_t)((1 - idx)* B_SZ + b) * 2 + 1) * HW);

    v8f acc_er = {};                      // WMMA C/D accumulators
    v8f acc_sm = {};
    v2f bones; bones[0] = 1.f; bones[1] = 1.f;   // B matrix = all ones

    const int trips = (NV4 + THREADS - 1) / THREADS;   // 13
    for (int t = 0; t < trips; ++t) {
        const int i = tid + t * THREADS;
        v2f a_er; a_er[0] = 0.f; a_er[1] = 0.f;
        v2f a_sm; a_sm[0] = 0.f; a_sm[1] = 0.f;
        if (i < NV4) {                    // predicated loads; WMMA stays full-EXEC
            const float4 v1 = c1[i];
            const float4 v2 = c2[i];
            const float dx = v1.x - v2.x, dy = v1.y - v2.y;
            const float dz = v1.z - v2.z, dw = v1.w - v2.w;
            a_er[0] = fmaf(dx, dx, dy * dy);
            a_er[1] = fmaf(dz, dz, dw * dw);
            if (need_same) {              // block-uniform: skip 3rd stream if masked
                const float4 vo = co[i];
                const float ex = v1.x - vo.x, ey = v1.y - vo.y;
                const float ez = v1.z - vo.z, ew = v1.w - vo.w;
                a_sm[0] = fmaf(ex, ex, ey * ey);
                a_sm[1] = fmaf(ez, ez, ew * ew);
            }
        }
        // D = A x ones + C : accumulate all 64 lane-values of A into row sums
        acc_er = __builtin_amdgcn_wmma_f32_16x16x4_f32(
                     false, a_er, false, bones, (short)0, acc_er, false, false);
        acc_sm = __builtin_amdgcn_wmma_f32_16x16x4_f32(
                     false, a_sm, false, bones, (short)0, acc_sm, false, false);
    }

    // Collapse D: per lane sum 8 VGPRs (rows M=0..7 or 8..15 of this half-wave),
    // then fold the two 16-lane halves together.
    float er_l = acc_er[0] + acc_er[1] + acc_er[2] + acc_er[3]
               + acc_er[4] + acc_er[5] + acc_er[6] + acc_er[7];
    float sm_l = acc_sm[0] + acc_sm[1] + acc_sm[2] + acc_sm[3]
               + acc_sm[4] + acc_sm[5] + acc_sm[6] + acc_sm[7];
    er_l += __shfl_xor(er_l, 16, 32);
    sm_l += __shfl_xor(sm_l, 16, 32);

    __shared__ float red_er[THREADS / 32];
    __shared__ float red_sm[THREADS / 32];
    const int wave = tid >> 5;
    const int lane = tid & 31;
    if (lane == 0) { red_er[wave] = er_l; red_sm[wave] = sm_l; }
    __syncthreads();
    if (tid == 0) {
        float s = 0.f, t2 = 0.f;
        for (int w = 0; w < THREADS / 32; ++w) { s += red_er[w]; t2 += red_sm[w]; }
        S_er[b]   = s;
        S_same[b] = t2;
    }
}

// Tiny per-sample scalar math (CE / softmax / argmax / weighting) + final mean.
__global__ __launch_bounds__(B_SZ)
void CombineLoss_finalize(const float* __restrict__ preds1,
                          const float* __restrict__ preds1_back,
                          const float* __restrict__ preds2,
                          const int*   __restrict__ y,
                          const int*   __restrict__ d_index,
                          const float* __restrict__ S_er,
                          const float* __restrict__ S_same,
                          float* __restrict__ out)
{
    __shared__ float red[B_SZ];
    const int b   = threadIdx.x;
    const int idx = (d_index[0] != 0) ? 1 : 0;

    const float p10 = preds1[((size_t)idx       * B_SZ + b) * 2 + 0];
    const float p11 = preds1[((size_t)idx       * B_SZ + b) * 2 + 1];
    const float q0  = preds1[((size_t)(1 - idx) * B_SZ + b) * 2 + 0];
    const float q1  = preds1[((size_t)(1 - idx) * B_SZ + b) * 2 + 1];
    const float r0  = preds2[((size_t)idx       * B_SZ + b) * 2 + 0];
    const float r1  = preds2[((size_t)idx       * B_SZ + b) * 2 + 1];
    const float pb0 = preds1_back[((size_t)idx  * B_SZ + b) * 2 + 0];
    const float pb1 = preds1_back[((size_t)idx  * B_SZ + b) * 2 + 1];
    const int   yb  = y[b];
    const float yf  = (float)yb;

    const float l1 = lse2(p10, p11);
    const float l2 = lse2(r0, r1);
    const float py1 = yb ? p11 : p10;
    const float py2 = yb ? r1 : r0;
    const float ce  = 0.5f * ((l1 - py1) + (l2 - py2));

    const float er      = yf * S_er[b] * (1.f / (float)HW);
    const float ce_back = 0.5f * (lse2(pb0, pb1) - pb0) * yf;   // BACK_RATIO = 1

    const int  cur  = (p11 > p10) ? 1 : 0;
    const int  flag = (q1  > q0 ) ? 1 : 0;
    const bool cond = (cur != flag) && (cur == 0) && (yb == 1);
    const float prob1 = expf(p11 - l1);              // softmax(p1)[1]
    const float w     = cond ? prob1 : 1.f;
    const float same  = (cur == flag) ? 1.f : 0.f;

    float contrib = w * (ce + er + ce_back) * (1.f / (float)B_SZ)
                  + yf * same * S_same[b] * (1.f / ((float)B_SZ * (float)HW));

    red[b] = contrib;
    __syncthreads();
    for (int s = B_SZ / 2; s > 0; s >>= 1) {
        if (b < s) red[b] += red[b + s];
        __syncthreads();
    }
    if (b == 0) out[0] = red[0];
}

extern "C" void kernel_launch(void* const* d_in, const int* in_sizes, int n_in,
                              void* d_out, int out_size, void* d_ws, size_t ws_size,
                              hipStream_t stream) {
    (void)in_sizes; (void)n_in; (void)out_size; (void)ws_size;
    const float* preds1      = (const float*)d_in[0];
    const float* cams1       = (const float*)d_in[1];
    const float* preds1_back = (const float*)d_in[2];
    const float* preds2      = (const float*)d_in[3];
    const float* cams2       = (const float*)d_in[4];
    const int*   y           = (const int*)d_in[5];
    const int*   index       = (const int*)d_in[6];

    float* S_er   = (float*)d_ws;
    float* S_same = S_er + B_SZ;

    CombineLoss_cam_sums<<<B_SZ, THREADS, 0, stream>>>(
        preds1, cams1, cams2, y, index, S_er, S_same);
    CombineLoss_finalize<<<1, B_SZ, 0, stream>>>(
        preds1, preds1_back, preds2, y, index, S_er, S_same, (float*)d_out);
}